// get_detector_14216341750325
// MI455X (gfx1250) — compile-verified
//
#include <hip/hip_runtime.h>
#include <stdint.h>

// ---------------- problem constants ----------------
#define NB     16
#define HH     128
#define WWID   128
#define NLOC   (HH*WWID)       // 16384
#define NCLS   80
#define NCC    (NLOC*NCLS)     // 1310720
#define TOPN_  1000
#define KPAD_  1024
#define NBINS  8192            // float bits >> 19 (exp + 4 mantissa bits)
#define CAP    4096            // candidate cap per batch for final sort
#define NMS_T  0.6f
#define CONF_T 0.05f
#define CLS_OFF 4096.0f

typedef __attribute__((ext_vector_type(2))) float v2f;
typedef __attribute__((ext_vector_type(8))) float v8f;

#if defined(__gfx1250__)
#define HAVE_ASYNC_LDS 1
// GLOBAL_LOAD_ASYNC_TO_LDS_B128 (VGLOBAL op 98): VDST = LDS byte offset,
// VADDR = 64-bit global address, SADDR = off. Tracked by ASYNCcnt.
// Low 32 bits of a flat shared-pointer are the LDS offset (aperture truncation).
__device__ __forceinline__ void async_g2lds_b128(const void* g, void* l) {
  unsigned lds_off = (unsigned)(uintptr_t)l;
  asm volatile("global_load_async_to_lds_b128 %0, %1, off"
               :: "v"(lds_off), "v"(g) : "memory");
}
__device__ __forceinline__ void async_wait0() {
  asm volatile("s_wait_asynccnt 0x0" ::: "memory");
}
#endif

__device__ __forceinline__ float sigf(float x) { return 1.0f / (1.0f + __expf(-x)); }

__device__ __forceinline__ unsigned ballot32(bool p) {
#if __has_builtin(__builtin_amdgcn_ballot_w32)
  return __builtin_amdgcn_ballot_w32(p);
#else
  return (unsigned)__ballot(p);
#endif
}

// ---------------- K0: zero scratch counters ----------------
__global__ void k_zero(unsigned* __restrict__ hist, unsigned* __restrict__ cc) {
  int i = blockIdx.x * blockDim.x + threadIdx.x;
  if (i < NB * NBINS) hist[i] = 0u;
  if (i < NB) cc[i] = 0u;
}

// ---------------- K1: score + radix histogram ----------------
__global__ void __launch_bounds__(256) k_hist(const float* __restrict__ cls,
                                              const float* __restrict__ ctr,
                                              unsigned* __restrict__ hist) {
  __shared__ unsigned lh[NBINS];
  const int b = blockIdx.y;
  for (int i = threadIdx.x; i < NBINS; i += blockDim.x) lh[i] = 0u;
  __syncthreads();
  const float* clsb = cls + (size_t)b * NCC;
  const float* ctrb = ctr + (size_t)b * NLOC;
  const int stride = gridDim.x * blockDim.x;
  for (int e = blockIdx.x * blockDim.x + threadIdx.x; e < NCC; e += stride) {
    __builtin_prefetch(clsb + e + 8 * stride, 0, 0);   // global_prefetch_b8
    float sc = sigf(clsb[e]);
    if (sc > CONF_T) {
      float s = sc * sigf(ctrb[e / NCLS]);
      unsigned key = __float_as_uint(s);               // s in (0,1): monotone bits
      if (key) atomicAdd(&lh[key >> 19], 1u);
    }
  }
  __syncthreads();
  for (int i = threadIdx.x; i < NBINS; i += blockDim.x)
    if (lh[i]) atomicAdd(&hist[b * NBINS + i], lh[i]);
}

// ---------------- K2: find threshold bin (radix select) ----------------
__global__ void k_thresh(const unsigned* __restrict__ hist, unsigned* __restrict__ thrBin) {
  int b = threadIdx.x;
  if (b < NB) {
    unsigned cum = 0; int t = 0;
    for (int i = NBINS - 1; i >= 0; --i) {
      cum += hist[b * NBINS + i];
      if (cum >= (unsigned)TOPN_) { t = i; break; }
    }
    thrBin[b] = (unsigned)t;
  }
}

// ---------------- K3: compact candidates >= threshold bin ----------------
__global__ void __launch_bounds__(256) k_collect(const float* __restrict__ cls,
                                                 const float* __restrict__ ctr,
                                                 const unsigned* __restrict__ thrBin,
                                                 unsigned* __restrict__ ccount,
                                                 unsigned* __restrict__ candKey,
                                                 unsigned* __restrict__ candIdx) {
  const int b = blockIdx.y;
  const unsigned tb = thrBin[b];
  const float* clsb = cls + (size_t)b * NCC;
  const float* ctrb = ctr + (size_t)b * NLOC;
  const int stride = gridDim.x * blockDim.x;
  for (int e = blockIdx.x * blockDim.x + threadIdx.x; e < NCC; e += stride) {
    float sc = sigf(clsb[e]);
    if (sc > CONF_T) {
      float s = sc * sigf(ctrb[e / NCLS]);
      unsigned key = __float_as_uint(s);
      if (key && (key >> 19) >= tb) {
        unsigned pos = atomicAdd(&ccount[b], 1u);
        if (pos < CAP) {
          candKey[(size_t)b * CAP + pos] = key;
          candIdx[(size_t)b * CAP + pos] = (unsigned)e;
        }
      }
    }
  }
}

// ---------------- K4: bitonic sort candidates + decode boxes ----------------
__global__ void __launch_bounds__(1024) k_sort_decode(
    const unsigned* __restrict__ ccount, const unsigned* __restrict__ candKey,
    const unsigned* __restrict__ candIdx, const float* __restrict__ boxes,
    const float* __restrict__ locations, const int* __restrict__ imh,
    const int* __restrict__ imw, float* __restrict__ det, float* __restrict__ obox,
    float* __restrict__ area, float* __restrict__ score, int* __restrict__ label,
    int* __restrict__ valid) {
  __shared__ unsigned skey[CAP];
  __shared__ unsigned sidx[CAP];
  const int b = blockIdx.x, tid = threadIdx.x, nt = blockDim.x;
  unsigned cnt = ccount[b]; if (cnt > CAP) cnt = CAP;
#ifdef HAVE_ASYNC_LDS
  // Async-stage both candidate arrays into LDS (ASYNCcnt path), then zero the tail.
  // nt == 1024, CAP == 4096 words: every thread issues exactly one b128 per array.
  {
    int i = tid * 4;
    async_g2lds_b128(candKey + (size_t)b * CAP + i, skey + i);
    async_g2lds_b128(candIdx + (size_t)b * CAP + i, sidx + i);
    async_wait0();
  }
  __syncthreads();
  for (int i = tid; i < CAP; i += nt)
    if (i >= (int)cnt) { skey[i] = 0u; sidx[i] = 0u; }
#else
  for (int i = tid; i < CAP; i += nt) {
    skey[i] = (i < (int)cnt) ? candKey[(size_t)b * CAP + i] : 0u;
    sidx[i] = (i < (int)cnt) ? candIdx[(size_t)b * CAP + i] : 0u;
  }
#endif
  __syncthreads();
  for (unsigned size = 2; size <= CAP; size <<= 1) {
    for (unsigned str = size >> 1; str > 0; str >>= 1) {
      for (unsigned k = tid; k < CAP; k += nt) {
        unsigned l = k ^ str;
        if (l > k) {
          bool desc = ((k & size) == 0);
          unsigned a = skey[k], c = skey[l];
          if (desc ? (a < c) : (a > c)) {
            skey[k] = c; skey[l] = a;
            unsigned t2 = sidx[k]; sidx[k] = sidx[l]; sidx[l] = t2;
          }
        }
      }
      __syncthreads();
    }
  }
  const float fw = (float)(*imw) - 1.0f, fh = (float)(*imh) - 1.0f;
  for (int t = tid; t < KPAD_; t += nt) {
    unsigned key = skey[t];
    int v = (t < TOPN_) && (key != 0u);
    unsigned idx = sidx[t];
    int loc = (int)(idx / NCLS);
    int lab = (int)(idx % NCLS) + 1;
    float4 pb = ((const float4*)boxes)[(size_t)b * NLOC + loc];
    float2 lc = ((const float2*)locations)[loc];
    float x1 = fminf(fmaxf(lc.x - pb.x, 0.f), fw);
    float y1 = fminf(fmaxf(lc.y - pb.y, 0.f), fh);
    float x2 = fminf(fmaxf(lc.x + pb.z, 0.f), fw);
    float y2 = fminf(fmaxf(lc.y + pb.w, 0.f), fh);
    float s = __uint_as_float(key);
    if (!v) { x1 = y1 = x2 = y2 = 0.f; lab = 0; s = 0.f; }
    size_t r = ((size_t)b * KPAD_ + t) * 4;
    det[r + 0] = x1; det[r + 1] = y1; det[r + 2] = x2; det[r + 3] = y2;
    float co = (float)lab * CLS_OFF;
    obox[r + 0] = x1 + co; obox[r + 1] = y1 + co;
    obox[r + 2] = x2 + co; obox[r + 3] = y2 + co;
    area[(size_t)b * KPAD_ + t] = fmaxf(x2 - x1, 0.f) * fmaxf(y2 - y1, 0.f);
    score[(size_t)b * KPAD_ + t] = s;
    label[(size_t)b * KPAD_ + t] = lab * v;
    valid[(size_t)b * KPAD_ + t] = v;
  }
}

// ---------------- K5: suppression bitmask via WMMA 16x16 IoU tiles -----------
// One wave32 owns a 16-row tile and sweeps all 64 column tiles.
// v_wmma_f32_16x16x4_f32 computes pairwise (area_i + area_j) per tile.
__global__ void __launch_bounds__(256) k_mask(const float* __restrict__ oboxf,
                                              const float* __restrict__ area,
                                              unsigned* __restrict__ mask) {
  const int b = blockIdx.y;
  const int wave = threadIdx.x >> 5, lane = threadIdx.x & 31;
  const int ti = blockIdx.x * 8 + wave;
  const int i0 = ti * 16;
  const float4* ob = (const float4*)oboxf + (size_t)b * KPAD_;
  const float* ar = area + (size_t)b * KPAD_;
  const int lhalf = lane & 15;
  const bool hi = lane >= 16;

  float4 bi[8];
#pragma unroll
  for (int v = 0; v < 8; ++v) bi[v] = ob[i0 + v + (hi ? 8 : 0)];

  // A (16x4 f32): lanes 0-15 supply K=0 (area_i) and K=1 (1.0); lanes 16-31 K=2,3 = 0
  v2f a; a.x = hi ? 0.f : ar[i0 + lhalf]; a.y = hi ? 0.f : 1.0f;

  unsigned loLo[8], loHi[8];
  for (int tj = 0; tj < 64; ++tj) {
    const int j0 = tj * 16;
    float4 bj = ob[j0 + lhalf];
    // B (4x16 f32): K=0 row = 1.0, K=1 row = area_j
    v2f bb; bb.x = hi ? 0.f : 1.0f; bb.y = hi ? 0.f : ar[j0 + lhalf];
    v8f c = {};
    c = __builtin_amdgcn_wmma_f32_16x16x4_f32(false, a, false, bb,
                                              (short)0, c, false, false);
    const int j = j0 + lhalf;
#pragma unroll
    for (int v = 0; v < 8; ++v) {
      const int i = i0 + v + (hi ? 8 : 0);
      float ix1 = fmaxf(bi[v].x, bj.x), iy1 = fmaxf(bi[v].y, bj.y);
      float ix2 = fminf(bi[v].z, bj.z), iy2 = fminf(bi[v].w, bj.w);
      float inter = fmaxf(ix2 - ix1, 0.f) * fmaxf(iy2 - iy1, 0.f);
      float iou = inter / (c[v] - inter + 1e-9f);          // c[v] = area_i+area_j
      bool pred = (iou > NMS_T) && (j > i);
      unsigned bal = ballot32(pred);   // low16: rows v, high16: rows v+8
      if ((tj & 1) == 0) {
        loLo[v] = bal & 0xFFFFu;
        loHi[v] = bal >> 16;
      } else {
        unsigned wLo = loLo[v] | ((bal & 0xFFFFu) << 16);
        unsigned wHi = loHi[v] | ((bal >> 16) << 16);
        if (lane == 0) {
          mask[((size_t)b * KPAD_ + i0 + v) * 32 + (tj >> 1)] = wLo;
          mask[((size_t)b * KPAD_ + i0 + v + 8) * 32 + (tj >> 1)] = wHi;
        }
      }
    }
  }
}

// ---------------- K6: exact greedy NMS scan (LDS-staged) + outputs ----------
__global__ void __launch_bounds__(1024) k_nms_out(const unsigned* __restrict__ mask,
                                                  const float* __restrict__ det,
                                                  const float* __restrict__ score,
                                                  const int* __restrict__ label,
                                                  const int* __restrict__ valid,
                                                  float* __restrict__ out) {
  extern __shared__ unsigned sm[];
  unsigned* sh_mask = sm;          // 32768 words
  unsigned* sh_keep = sm + 32768;  // 32 words
  const int b = blockIdx.x, tid = threadIdx.x;
#ifdef HAVE_ASYNC_LDS
  {
    const unsigned* gsrc = mask + (size_t)b * KPAD_ * 32;
    // 32768 words / (1024 threads * 4 words) = 8 uniform iterations per thread
    for (int i = tid * 4; i < KPAD_ * 32; i += blockDim.x * 4)
      async_g2lds_b128(gsrc + i, sh_mask + i);
    async_wait0();
  }
#else
  for (int i = tid; i < KPAD_ * 32; i += blockDim.x)
    sh_mask[i] = mask[(size_t)b * KPAD_ * 32 + i];
#endif
  __syncthreads();
  if (tid < 32) {
    const int lane = tid;
    unsigned keep = 0u;
    for (int t = 0; t < 32; ++t)
      if (valid[(size_t)b * KPAD_ + lane * 32 + t]) keep |= (1u << t);
    for (int i = 0; i < KPAD_; ++i) {
      unsigned m = sh_mask[i * 32 + lane];
      unsigned kw = (unsigned)__shfl((int)keep, i >> 5, 32);
      if ((kw >> (i & 31)) & 1u) keep &= ~m;
    }
    sh_keep[lane] = keep;
  }
  __syncthreads();
  const int i = tid;
  if (i < TOPN_) {
    float k = ((sh_keep[i >> 5] >> (i & 31)) & 1u) ? 1.f : 0.f;
    const float* d = det + ((size_t)b * KPAD_ + i) * 4;
    float* bo = out + ((size_t)b * TOPN_ + i) * 4;
    bo[0] = d[0] * k; bo[1] = d[1] * k; bo[2] = d[2] * k; bo[3] = d[3] * k;
    out[(size_t)NB * TOPN_ * 4 + (size_t)b * TOPN_ + i] =
        (k != 0.f) ? score[(size_t)b * KPAD_ + i] : 0.f;
    ((int*)out)[(size_t)NB * TOPN_ * 5 + (size_t)b * TOPN_ + i] =
        (k != 0.f) ? label[(size_t)b * KPAD_ + i] : 0;
  }
}

// ---------------- host launcher ----------------
extern "C" void kernel_launch(void* const* d_in, const int* in_sizes, int n_in,
                              void* d_out, int out_size, void* d_ws, size_t ws_size,
                              hipStream_t stream) {
  (void)in_sizes; (void)n_in; (void)out_size; (void)ws_size;
  const float* locations = (const float*)d_in[0];
  const float* pred_cls  = (const float*)d_in[1];
  const float* pred_box  = (const float*)d_in[2];
  const float* pred_ctr  = (const float*)d_in[3];
  const int*   imh       = (const int*)d_in[4];
  const int*   imw       = (const int*)d_in[5];

  char* ws = (char*)d_ws;
  size_t off = 0;
  auto carve = [&](size_t bytes) -> char* {
    char* p = ws + off;
    off = (off + bytes + 255) & ~(size_t)255;
    return p;
  };
  unsigned* hist    = (unsigned*)carve((size_t)NB * NBINS * 4);
  unsigned* ccount  = (unsigned*)carve((size_t)NB * 4);
  unsigned* thrBin  = (unsigned*)carve((size_t)NB * 4);
  unsigned* candKey = (unsigned*)carve((size_t)NB * CAP * 4);
  unsigned* candIdx = (unsigned*)carve((size_t)NB * CAP * 4);
  float*    det     = (float*)carve((size_t)NB * KPAD_ * 4 * 4);
  float*    obox    = (float*)carve((size_t)NB * KPAD_ * 4 * 4);
  float*    area    = (float*)carve((size_t)NB * KPAD_ * 4);
  float*    score   = (float*)carve((size_t)NB * KPAD_ * 4);
  int*      label   = (int*)carve((size_t)NB * KPAD_ * 4);
  int*      valid   = (int*)carve((size_t)NB * KPAD_ * 4);
  unsigned* supmask = (unsigned*)carve((size_t)NB * KPAD_ * 32 * 4);

  k_zero<<<(NB * NBINS + 255) / 256, 256, 0, stream>>>(hist, ccount);
  k_hist<<<dim3(128, NB), 256, 0, stream>>>(pred_cls, pred_ctr, hist);
  k_thresh<<<1, NB, 0, stream>>>(hist, thrBin);
  k_collect<<<dim3(128, NB), 256, 0, stream>>>(pred_cls, pred_ctr, thrBin,
                                               ccount, candKey, candIdx);
  k_sort_decode<<<NB, 1024, 0, stream>>>(ccount, candKey, candIdx, pred_box,
                                         locations, imh, imw, det, obox, area,
                                         score, label, valid);
  k_mask<<<dim3(8, NB), 256, 0, stream>>>(obox, area, supmask);
  k_nms_out<<<NB, 1024, (KPAD_ * 32 + 32) * sizeof(unsigned), stream>>>(
      supmask, det, score, label, valid, (float*)d_out);
}